// Kmer_36283883717364
// MI455X (gfx1250) — compile-verified
//
#include <hip/hip_runtime.h>
#include <cstdint>

// K-mer histogram for MI455X (gfx1250):
//   - full 65536-bin f32 histogram resident in LDS (256 KB of the 320 KB/WGP)
//   - ds_add_f32 LDS atomics for the scatter
//   - async LDS->global b128 stores (ASYNCcnt path) for the 64 MB writeout

#define TPB 1024

constexpr int B_    = 256;
constexpr int L_    = 8192;
constexpr int K_    = 8;
constexpr int LO_   = L_ - K_ + 1;          // 8185 valid k-mers per row
constexpr int NBINS = 65536;                // 4^8
constexpr int HIST_BYTES = NBINS * 4;       // 262144
constexpr int SYM_BYTES  = L_ + 16;         // 8192 symbols + pad for b64 tail reads
constexpr int SMEM_BYTES = HIST_BYTES + SYM_BYTES;  // 270352 <= 320 KB/WGP

__global__ __launch_bounds__(TPB, 1)
void Kmer_36283883717364_kernel(const float* __restrict__ seq,
                                float* __restrict__ out) {
    extern __shared__ unsigned char smem[];
    float*         hist = reinterpret_cast<float*>(smem);
    unsigned char* sym  = smem + HIST_BYTES;

    const int b = blockIdx.x;
    const int t = threadIdx.x;

    // ---- zero the LDS histogram (ds_store_b128, 16 B/lane/iter) ----
    float4  z4 = make_float4(0.f, 0.f, 0.f, 0.f);
    float4* h4 = reinterpret_cast<float4*>(hist);
#pragma unroll
    for (int k = 0; k < NBINS / 4 / TPB; ++k)      // 16 iters
        h4[t + k * TPB] = z4;
    if (t < 16) sym[L_ + t] = 0;                   // pad tail

    // ---- phase 1: argmax over 4 logits -> one symbol byte per position ----
    // Coalesced float4 loads: 512 B per wave32 per iteration.
    const float4* row = reinterpret_cast<const float4*>(seq) + (size_t)b * L_;
#pragma unroll
    for (int k = 0; k < L_ / TPB; ++k) {           // 8 iters
        int    p = t + k * TPB;
        float4 x = row[p];
        int m = 0; float v = x.x;                  // strict '>' => first-max,
        if (x.y > v) { v = x.y; m = 1; }           // matching jnp.argmax ties
        if (x.z > v) { v = x.z; m = 2; }
        if (x.w > v) { v = x.w; m = 3; }
        sym[p] = (unsigned char)m;
    }
    __syncthreads();

    // ---- phase 2: rolling 8-mer codes + LDS f32 atomic histogram ----
    const int i0 = t * 8;                          // 8 k-mers per thread
    if (i0 < LO_) {
        const int count = (LO_ - i0 < 8) ? (LO_ - i0) : 8;
        // two ds_load_b64: symbol bytes i0 .. i0+15 (pad keeps this in range)
        const uint64_t* sp = reinterpret_cast<const uint64_t*>(sym + i0);
        uint64_t s0 = sp[0];
        uint64_t s1 = sp[1];
        unsigned code = 0;
#pragma unroll
        for (int j = 0; j < 7; ++j)                // prime first 7 symbols
            code = (code << 2) | (unsigned)((s0 >> (8 * j)) & 3ull);
#pragma unroll
        for (int c = 0; c < 8; ++c) {
            if (c < count) {
                unsigned nb = (c == 0)
                    ? (unsigned)((s0 >> 56) & 3ull)
                    : (unsigned)((s1 >> (8 * (c - 1))) & 3ull);
                code = ((code << 2) | nb) & 0xFFFFu;
                atomicAdd(&hist[code], 1.0f);      // -> ds_add_f32 (no rtn)
            }
        }
    }
    __syncthreads();

    // ---- phase 3: async LDS -> global writeout (ASYNCcnt path) ----
    // Histogram already holds f32 counts, so this is a pure 256 KB copy:
    // global_store_async_from_lds_b128, 16 B per lane, 16 rounds.
    unsigned char* orow =
        reinterpret_cast<unsigned char*>(out) + (size_t)b * HIST_BYTES;
#pragma unroll
    for (int k = 0; k < HIST_BYTES / (TPB * 16); ++k) {   // 16 iters
        unsigned byteoff = (unsigned)(t + k * TPB) * 16u;
        // low 32 bits of a generic shared pointer == LDS byte offset
        unsigned lds_off = (unsigned)(uintptr_t)(smem + byteoff);
        uint64_t gaddr   = (uint64_t)(uintptr_t)(orow + byteoff);
        asm volatile("global_store_async_from_lds_b128 %0, %1, off"
                     :
                     : "v"(gaddr), "v"(lds_off)
                     : "memory");
    }
    asm volatile("s_wait_asynccnt 0" ::: "memory");
}

extern "C" void kernel_launch(void* const* d_in, const int* in_sizes, int n_in,
                              void* d_out, int out_size, void* d_ws, size_t ws_size,
                              hipStream_t stream) {
    (void)in_sizes; (void)n_in; (void)d_ws; (void)ws_size; (void)out_size;

    const float* seq = (const float*)d_in[0];   // [256, 8192, 4] f32
    float*       out = (float*)d_out;           // [256, 65536] f32

    // Allow >64 KB dynamic LDS (host-side attribute; graph-capture safe).
    hipFuncSetAttribute((const void*)Kmer_36283883717364_kernel,
                        hipFuncAttributeMaxDynamicSharedMemorySize, SMEM_BYTES);

    Kmer_36283883717364_kernel<<<dim3(B_), dim3(TPB), SMEM_BYTES, stream>>>(seq, out);
}